// ts_corr_14362370638088
// MI455X (gfx1250) — compile-verified
//
#include <hip/hip_runtime.h>

typedef __attribute__((ext_vector_type(2))) float v2f;
typedef __attribute__((ext_vector_type(4))) float v4f;
typedef __attribute__((ext_vector_type(8))) float v8f;

#define WINDOW  5
#define FDIM    16
#define TTOT    256
#define SLEN    252      // number of window starts
#define NPAIR   120      // 16*15/2 upper-triangular pairs
#define SCHUNK  32       // s-values per block
#define LDSP    36       // padded LDS row stride (floats): x4 for b128, offset vs 32 for banks
#define EPSV    1e-5f
#define RSQRTW  0.44721359549995793f   // 1/sqrt(5)

__global__ __launch_bounds__(256)
void ts_corr_wmma_kernel(const float* __restrict__ in, float* __restrict__ out)
{
    const int b      = blockIdx.y;
    const int s_base = blockIdx.x * SCHUNK;
    const int tid    = threadIdx.x;
    const int lane   = tid & 31;
    const int wave   = tid >> 5;
    const int f      = lane & 15;   // feature column this lane owns
    const int half   = lane >> 4;   // K-half of the WMMA operand this lane holds

    // 120x32 staging (stride 36, 16B-aligned rows) + dump region for invalid slots
    __shared__ __align__(16) float lds[NPAIR * LDSP + 32];

    // Hoisted scatter addresses: slot (r,lane) -> LDS row base; tile only adds +j.
    // Valid (m<f) slots map to triu row p; invalid ones land in the dump region.
    int abase[8];
    #pragma unroll
    for (int r = 0; r < 8; ++r) {
        const int m = r + 8 * half;
        const int p = m * (31 - m) / 2 + (f - m - 1);      // valid iff m < f
        abase[r] = (m < f) ? (p * LDSP) : (NPAIR * LDSP);  // dump row + j stays in-bounds
    }

    const float* bin = in + (size_t)b * TTOT * FDIM;

    // This wave's 4 consecutive windows need 8 consecutive rows: load once.
    const int s0 = s_base + wave * 4;
    float xr[8];
    #pragma unroll
    for (int k = 0; k < 8; ++k) {
        int row = s0 + k;
        row = row < TTOT ? row : TTOT - 1;     // clamp (only hits masked tail tiles)
        xr[k] = bin[row * FDIM + f];
    }

    #pragma unroll
    for (int t = 0; t < 4; ++t) {
        const int j = wave * 4 + t;            // position within s-chunk: 0..31
        float x0 = xr[t], x1 = xr[t+1], x2 = xr[t+2], x3 = xr[t+3], x4 = xr[t+4];

        // Per-column mean / std (ddof=1), then z-normalize so WMMA emits corr directly:
        //   y_w = (x_w - mean) / (std * sqrt(W))  =>  corr[i][j] = sum_w y_wi * y_wj
        float mean  = (x0 + x1 + x2 + x3 + x4) * 0.2f;
        float sq    = x0*x0 + x1*x1 + x2*x2 + x3*x3 + x4*x4;
        float var   = (sq - 5.0f * mean * mean) * 0.25f;
        float stdev = __builtin_amdgcn_sqrtf(var) + EPSV;
        float scale = __builtin_amdgcn_rcpf(stdev) * RSQRTW;

        float y0 = (x0 - mean) * scale;
        float y1 = (x1 - mean) * scale;
        float y2 = (x2 - mean) * scale;
        float y3 = (x3 - mean) * scale;
        float y4 = (x4 - mean) * scale;

        // f32 16x16x4 operand layout: lanes 0-15 hold K={0,1}, lanes 16-31 K={2,3}.
        // A (=Y^T, 16x4) and B (=Y, 4x16) register images coincide for Gram product.
        v2f ab0, ab1;
        ab0.x = half ? y2 : y0;      // K-chunk 0: K = 0..3
        ab0.y = half ? y3 : y1;
        ab1.x = half ? 0.0f : y4;    // K-chunk 1: K = 4..7 (only K=4 real)
        ab1.y = 0.0f;

        v8f c = {};
        c = __builtin_amdgcn_wmma_f32_16x16x4_f32(false, ab0, false, ab0,
                                                  (short)0, c, false, false);
        c = __builtin_amdgcn_wmma_f32_16x16x4_f32(false, ab1, false, ab1,
                                                  (short)0, c, false, false);

        // C/D layout: VGPR r, lanes 0-15 -> (M=r, N=lane); lanes 16-31 -> (M=r+8).
        // Branchless scatter using precomputed row bases.
        #pragma unroll
        for (int r = 0; r < 8; ++r)
            lds[abase[r] + j] = c[r];
    }

    __syncthreads();

    // B128 writeback: 120 rows x 8 float4s along s. All addresses 16B-aligned
    // (row stride 1008B, chunk base 128B, tail s-count 28 = 4*7 — float4 exact).
    const size_t obase = (size_t)b * NPAIR * SLEN;
    for (int idx = tid; idx < NPAIR * (SCHUNK / 4); idx += 256) {
        const int p  = idx >> 3;
        const int jj = idx & 7;
        const int s  = s_base + 4 * jj;
        if (s < SLEN) {
            v4f v = *(const v4f*)&lds[p * LDSP + 4 * jj];
            *(v4f*)&out[obase + (size_t)p * SLEN + s] = v;
        }
    }
}

extern "C" void kernel_launch(void* const* d_in, const int* in_sizes, int n_in,
                              void* d_out, int out_size, void* d_ws, size_t ws_size,
                              hipStream_t stream)
{
    const float* in = (const float*)d_in[0];
    float* out      = (float*)d_out;

    // gridDim.x: 8 s-chunks of 32 (covers 252 with masked tail); gridDim.y: 1024 batches
    dim3 grid((SLEN + SCHUNK - 1) / SCHUNK, 1024);
    dim3 block(256);
    ts_corr_wmma_kernel<<<grid, block, 0, stream>>>(in, out);
}